// FDModel_80384607912254
// MI455X (gfx1250) — compile-verified
//
#include <hip/hip_runtime.h>
#include <hip/hip_bf16.h>

#define N_  4096
#define H_  256
#define D_  512
#define V_  3
#define C_  20
#define DY_ 300
#define LDP 36   // LDS row pitch (floats): keeps 16B-aligned b128 stores, conflict-free b64 reads

typedef float v2f __attribute__((ext_vector_type(2)));
typedef float v8f __attribute__((ext_vector_type(8)));

static __device__ __forceinline__ v8f wmma_f32_16x16x4(v2f a, v2f b, v8f c) {
  // D = A(16x4) * B(4x16) + C, exact fp32 path on gfx1250
  return __builtin_amdgcn_wmma_f32_16x16x4_f32(false, a, false, b, (short)0, c, false, false);
}

// ---------------- Kernel 1: xp = leaky(x @ W^T + b) -------------------------
// Tile: 128 (rows n) x 64 (cols h), K = 512. 8 waves, each wave = 16x64 strip.
__global__ __launch_bounds__(256) void proj_kernel(const float* __restrict__ X,
                                                   const float* __restrict__ W,
                                                   const float* __restrict__ bias,
                                                   float* __restrict__ XPo) {
  __shared__ float As[128 * LDP];
  __shared__ float Bs[64 * LDP];
  const int tid  = threadIdx.x;
  const int lane = tid & 31;
  const int wave = tid >> 5;
  const int l15  = lane & 15;
  const int hl   = lane >> 4;
  const int n0 = blockIdx.x * 128;
  const int h0 = blockIdx.y * 64;

  v8f acc[4];
  #pragma unroll
  for (int t = 0; t < 4; ++t)
    #pragma unroll
    for (int j = 0; j < 8; ++j) acc[t][j] = 0.0f;

  const int arow = wave * 16 + l15;
  const int kb0  = hl * 2;

  for (int k0 = 0; k0 < D_; k0 += 32) {
    __syncthreads();
    for (int idx = tid; idx < 128 * 8; idx += 256) {
      const int r = idx >> 3, c4 = (idx & 7) << 2;
      *(float4*)(&As[r * LDP + c4]) =
          *(const float4*)(X + (size_t)(n0 + r) * D_ + k0 + c4);
    }
    for (int idx = tid; idx < 64 * 8; idx += 256) {
      const int r = idx >> 3, c4 = (idx & 7) << 2;
      *(float4*)(&Bs[r * LDP + c4]) =
          *(const float4*)(W + (size_t)(h0 + r) * D_ + k0 + c4);
    }
    __syncthreads();
    #pragma unroll
    for (int kk = 0; kk < 8; ++kk) {
      const int kb = kk * 4 + kb0;
      const v2f a = *(const v2f*)(&As[arow * LDP + kb]);
      #pragma unroll
      for (int t = 0; t < 4; ++t) {
        const v2f b = *(const v2f*)(&Bs[(t * 16 + l15) * LDP + kb]);
        acc[t] = wmma_f32_16x16x4(a, b, acc[t]);
      }
    }
  }

  #pragma unroll
  for (int t = 0; t < 4; ++t) {
    const int col = h0 + t * 16 + l15;
    const float bv = bias[col];
    #pragma unroll
    for (int i = 0; i < 8; ++i) {
      const int row = n0 + wave * 16 + hl * 8 + i;
      float vv = acc[t][i] + bv;
      vv = (vv >= 0.0f) ? vv : 0.1f * vv;
      XPo[(size_t)row * H_ + col] = vv;
    }
  }
}

// ---------------- Kernel 2: xq = xp / max(||xp||, 1e-12) --------------------
__global__ __launch_bounds__(256) void normalize_kernel(const float* __restrict__ XP,
                                                        float* __restrict__ XQ) {
  __shared__ float red[256];
  const int row = blockIdx.x;        // flat (v*N + n)
  const int tid = threadIdx.x;
  const float v = XP[(size_t)row * H_ + tid];
  red[tid] = v * v;
  __syncthreads();
  for (int s = 128; s > 0; s >>= 1) {
    if (tid < s) red[tid] += red[tid + s];
    __syncthreads();
  }
  const float nrm = fmaxf(sqrtf(red[0]), 1e-12f);
  XQ[(size_t)row * H_ + tid] = v / nrm;
}

// -------- Kernel 3: S[n,m] = max_v masked exp(5 * <xq_v[n],xq_v[m]>) --------
// Tile: 128 x 64, K = 256, loops the 3 views internally keeping a running max.
__global__ __launch_bounds__(256) void sim_kernel(const float* __restrict__ XQ,
                                                  const int* __restrict__ mask,
                                                  float* __restrict__ S) {
  __shared__ float As[128 * LDP];
  __shared__ float Bs[64 * LDP];
  const int tid  = threadIdx.x;
  const int lane = tid & 31;
  const int wave = tid >> 5;
  const int l15  = lane & 15;
  const int hl   = lane >> 4;
  const int n0 = blockIdx.x * 128;
  const int m0 = blockIdx.y * 64;
  const int arow = wave * 16 + l15;
  const int kb0  = hl * 2;

  v8f smax[4];
  #pragma unroll
  for (int t = 0; t < 4; ++t)
    #pragma unroll
    for (int j = 0; j < 8; ++j) smax[t][j] = 0.0f;

  for (int v = 0; v < V_; ++v) {
    const float* Xq = XQ + (size_t)v * N_ * H_;
    v8f acc[4];
    #pragma unroll
    for (int t = 0; t < 4; ++t)
      #pragma unroll
      for (int j = 0; j < 8; ++j) acc[t][j] = 0.0f;

    for (int k0 = 0; k0 < H_; k0 += 32) {
      __syncthreads();
      for (int idx = tid; idx < 128 * 8; idx += 256) {
        const int r = idx >> 3, c4 = (idx & 7) << 2;
        *(float4*)(&As[r * LDP + c4]) =
            *(const float4*)(Xq + (size_t)(n0 + r) * H_ + k0 + c4);
      }
      for (int idx = tid; idx < 64 * 8; idx += 256) {
        const int r = idx >> 3, c4 = (idx & 7) << 2;
        *(float4*)(&Bs[r * LDP + c4]) =
            *(const float4*)(Xq + (size_t)(m0 + r) * H_ + k0 + c4);
      }
      __syncthreads();
      #pragma unroll
      for (int kk = 0; kk < 8; ++kk) {
        const int kb = kk * 4 + kb0;
        const v2f a = *(const v2f*)(&As[arow * LDP + kb]);
        #pragma unroll
        for (int t = 0; t < 4; ++t) {
          const v2f b = *(const v2f*)(&Bs[(t * 16 + l15) * LDP + kb]);
          acc[t] = wmma_f32_16x16x4(a, b, acc[t]);
        }
      }
    }
    // masked exp + running max over views
    #pragma unroll
    for (int t = 0; t < 4; ++t) {
      const int col = m0 + t * 16 + l15;
      const int mc = mask[col * V_ + v];
      #pragma unroll
      for (int i = 0; i < 8; ++i) {
        const int row = n0 + wave * 16 + hl * 8 + i;
        const int mr = mask[row * V_ + v];
        const float e = __expf(acc[t][i] * 5.0f);   // 1/BETA = 5
        const float sv = (mr != 0 && mc != 0) ? e : 0.0f;
        smax[t][i] = fmaxf(smax[t][i], sv);
      }
    }
  }

  #pragma unroll
  for (int t = 0; t < 4; ++t) {
    const int col = m0 + t * 16 + l15;
    #pragma unroll
    for (int i = 0; i < 8; ++i) {
      const int row = n0 + wave * 16 + hl * 8 + i;
      float sv = smax[t][i];
      if (row == col) sv = 0.0f;                    // zero diagonal
      S[(size_t)row * N_ + col] = sv;
    }
  }
}

// -------- Kernel 4: row sums + masked row-max -> rowsum, confi --------------
__global__ __launch_bounds__(256) void rowreduce_kernel(const float* __restrict__ S,
                                                        const int* __restrict__ mask,
                                                        float* __restrict__ rowsum,
                                                        float* __restrict__ confi) {
  __shared__ float rsum[256];
  __shared__ float rmax[3][256];
  const int n = blockIdx.x, tid = threadIdx.x;
  float sum = 0.0f, m0v = 0.0f, m1v = 0.0f, m2v = 0.0f;
  for (int m = tid; m < N_; m += 256) {
    const float sv = S[(size_t)n * N_ + m];
    sum += sv;
    if (mask[m * V_ + 0]) m0v = fmaxf(m0v, sv);
    if (mask[m * V_ + 1]) m1v = fmaxf(m1v, sv);
    if (mask[m * V_ + 2]) m2v = fmaxf(m2v, sv);
  }
  rsum[tid] = sum; rmax[0][tid] = m0v; rmax[1][tid] = m1v; rmax[2][tid] = m2v;
  __syncthreads();
  for (int s = 128; s > 0; s >>= 1) {
    if (tid < s) {
      rsum[tid] += rsum[tid + s];
      rmax[0][tid] = fmaxf(rmax[0][tid], rmax[0][tid + s]);
      rmax[1][tid] = fmaxf(rmax[1][tid], rmax[1][tid + s]);
      rmax[2][tid] = fmaxf(rmax[2][tid], rmax[2][tid + s]);
    }
    __syncthreads();
  }
  if (tid == 0) {
    rowsum[n] = rsum[0];
    #pragma unroll
    for (int v = 0; v < V_; ++v) {
      // max_m log(s)*BETA over masked m == BETA*log(masked row-max) (log monotone)
      float cv = 0.2f * logf(rmax[v][0] + 1e-9f);
      cv = fminf(fmaxf(cv, 0.0f), 1.0f);
      confi[(size_t)v * N_ + n] = cv;
    }
  }
}

// -------- Kernel 5: y_n = sigmoid(y @ Wy^T + by) (tiny: 20x256, K=300) ------
__global__ __launch_bounds__(256) void yn_kernel(const float* __restrict__ Y,
                                                 const float* __restrict__ Wy,
                                                 const float* __restrict__ by,
                                                 float* __restrict__ YN) {
  const int c = blockIdx.x, h = threadIdx.x;
  float acc = by[h];
  const float* yr = Y + (size_t)c * DY_;
  const float* wr = Wy + (size_t)h * DY_;
  for (int d = 0; d < DY_; ++d) acc += yr[d] * wr[d];
  YN[c * H_ + h] = 1.0f / (1.0f + __expf(-acc));
}

// -------- Kernel 6: new_x[v] = (S/rowsum) @ (mask_v * xp[v]); blend ---------
// Tile 128 x 64, K = 4096; B staged transposed in LDS with column mask folded in.
__global__ __launch_bounds__(256) void att_kernel(const float* __restrict__ S,
                                                  const float* __restrict__ XP,
                                                  const int* __restrict__ mask,
                                                  const float* __restrict__ rowsum,
                                                  float* __restrict__ OUTX) {
  __shared__ float As[128 * LDP];
  __shared__ float Bst[64 * LDP];   // [h][k] so WMMA b-operand is a b64 load
  const int tid  = threadIdx.x;
  const int lane = tid & 31;
  const int wave = tid >> 5;
  const int l15  = lane & 15;
  const int hl   = lane >> 4;
  const int v  = blockIdx.z;
  const int n0 = blockIdx.x * 128;
  const int h0 = blockIdx.y * 64;
  const float* XPv = XP + (size_t)v * N_ * H_;
  const int arow = wave * 16 + l15;
  const int kb0  = hl * 2;

  v8f acc[4];
  #pragma unroll
  for (int t = 0; t < 4; ++t)
    #pragma unroll
    for (int j = 0; j < 8; ++j) acc[t][j] = 0.0f;

  for (int k0 = 0; k0 < N_; k0 += 32) {
    __syncthreads();
    for (int idx = tid; idx < 128 * 8; idx += 256) {
      const int r = idx >> 3, c4 = (idx & 7) << 2;
      *(float4*)(&As[r * LDP + c4]) =
          *(const float4*)(S + (size_t)(n0 + r) * N_ + k0 + c4);
    }
    for (int idx = tid; idx < 32 * 16; idx += 256) {
      const int kr = idx >> 4;           // 0..31 (K within stage)
      const int h4 = (idx & 15) << 2;    // 0..60 (h within tile)
      const float4 w = *(const float4*)(XPv + (size_t)(k0 + kr) * H_ + h0 + h4);
      const float mk = (mask[(k0 + kr) * V_ + v] != 0) ? 1.0f : 0.0f;
      Bst[(h4 + 0) * LDP + kr] = w.x * mk;
      Bst[(h4 + 1) * LDP + kr] = w.y * mk;
      Bst[(h4 + 2) * LDP + kr] = w.z * mk;
      Bst[(h4 + 3) * LDP + kr] = w.w * mk;
    }
    __syncthreads();
    #pragma unroll
    for (int kk = 0; kk < 8; ++kk) {
      const int kb = kk * 4 + kb0;
      const v2f a = *(const v2f*)(&As[arow * LDP + kb]);
      #pragma unroll
      for (int t = 0; t < 4; ++t) {
        const v2f b = *(const v2f*)(&Bst[(t * 16 + l15) * LDP + kb]);
        acc[t] = wmma_f32_16x16x4(a, b, acc[t]);
      }
    }
  }

  #pragma unroll
  for (int t = 0; t < 4; ++t) {
    const int col = h0 + t * 16 + l15;
    #pragma unroll
    for (int i = 0; i < 8; ++i) {
      const int row = n0 + wave * 16 + hl * 8 + i;
      float val = acc[t][i] / (rowsum[row] + 1e-9f);
      if (mask[row * V_ + v] != 0) val = XPv[(size_t)row * H_ + col];  // blend
      OUTX[(size_t)v * N_ * H_ + (size_t)row * H_ + col] = val;
    }
  }
}

// -------- Kernel 7: Z[v,n,c,h] = new_x[v,n,h] * y_n[c,h] (252 MB write) -----
__global__ __launch_bounds__(256) void zout_kernel(const float* __restrict__ NX,
                                                   const float* __restrict__ YN,
                                                   float* __restrict__ Z) {
  const int n = blockIdx.x, v = blockIdx.y, h = threadIdx.x;
  const float xv = NX[((size_t)v * N_ + n) * H_ + h];
  float* zp = Z + ((size_t)v * N_ + n) * (size_t)(C_ * H_) + h;
  #pragma unroll
  for (int c = 0; c < C_; ++c) zp[(size_t)c * H_] = xv * YN[c * H_ + h];
}

extern "C" void kernel_launch(void* const* d_in, const int* in_sizes, int n_in,
                              void* d_out, int out_size, void* d_ws, size_t ws_size,
                              hipStream_t stream) {
  (void)in_sizes; (void)n_in; (void)out_size; (void)ws_size;
  // setup_inputs dict order: x0,W0,b0, x1,W1,b1, x2,W2,b2, y,Wy,by, mask
  const float* x[V_]  = {(const float*)d_in[0], (const float*)d_in[3], (const float*)d_in[6]};
  const float* Wv[V_] = {(const float*)d_in[1], (const float*)d_in[4], (const float*)d_in[7]};
  const float* bv[V_] = {(const float*)d_in[2], (const float*)d_in[5], (const float*)d_in[8]};
  const float* Y    = (const float*)d_in[9];
  const float* Wy   = (const float*)d_in[10];
  const float* by   = (const float*)d_in[11];
  const int*   mask = (const int*)d_in[12];

  float* ws = (float*)d_ws;
  const size_t VNH = (size_t)V_ * N_ * H_;
  float* XP     = ws;                       // V*N*H
  float* XQ     = ws + VNH;                 // V*N*H (reused as NEWXB after sim)
  float* S      = ws + 2 * VNH;             // N*N  (64 MB, fits global L2)
  float* ROWSUM = S + (size_t)N_ * N_;      // N
  float* YN     = ROWSUM + N_;              // C*H
  float* NEWXB  = XQ;                       // xq dead after sim_kernel

  float* Z     = (float*)d_out;
  float* CONFI = Z + (size_t)V_ * N_ * C_ * H_;

  for (int v = 0; v < V_; ++v)
    proj_kernel<<<dim3(N_ / 128, H_ / 64), 256, 0, stream>>>(
        x[v], Wv[v], bv[v], XP + (size_t)v * N_ * H_);
  normalize_kernel<<<V_ * N_, 256, 0, stream>>>(XP, XQ);
  sim_kernel<<<dim3(N_ / 128, N_ / 64), 256, 0, stream>>>(XQ, mask, S);
  rowreduce_kernel<<<N_, 256, 0, stream>>>(S, mask, ROWSUM, CONFI);
  yn_kernel<<<C_, 256, 0, stream>>>(Y, Wy, by, YN);
  att_kernel<<<dim3(N_ / 128, H_ / 64, V_), 256, 0, stream>>>(S, XP, mask, ROWSUM, NEWXB);
  zout_kernel<<<dim3(N_, V_), 256, 0, stream>>>(NEWXB, YN, Z);
}